// Ensemble_KF_low_36876589203566
// MI455X (gfx1250) — compile-verified
//
#include <hip/hip_runtime.h>
#include <hip/hip_bf16.h>

typedef __attribute__((ext_vector_type(2))) float v2f;
typedef __attribute__((ext_vector_type(8))) float v8f;

#define B_ 512
#define K_ 32
#define DX_ 64
#define DZ_ 32
#define DA_ 32
#define RAW_ 30
#define BK_ (B_ * K_)

// 16-wide k-block of operands for a 32x32 output tile (2 A tiles, 2 B tiles)
struct Blk {
    v2f a0[4], a1[4], b0[4], b1[4];
};

__device__ __forceinline__ void load_blk(Blk& t,
    const float* __restrict__ xr0, const float* __restrict__ xr1,
    const float* __restrict__ wr0, const float* __restrict__ wr1, int k0)
{
#pragma unroll
    for (int kk = 0; kk < 4; kk++) {
        t.a0[kk] = *(const v2f*)(xr0 + k0 + 4 * kk);
        t.a1[kk] = *(const v2f*)(xr1 + k0 + 4 * kk);
        t.b0[kk] = *(const v2f*)(wr0 + k0 + 4 * kk);
        t.b1[kk] = *(const v2f*)(wr1 + k0 + 4 * kk);
    }
}

__device__ __forceinline__ void mma_blk(const Blk& t,
    v8f& c00, v8f& c01, v8f& c10, v8f& c11)
{
#pragma unroll
    for (int kk = 0; kk < 4; kk++) {
        c00 = __builtin_amdgcn_wmma_f32_16x16x4_f32(false, t.a0[kk], false, t.b0[kk], (short)0, c00, false, false);
        c01 = __builtin_amdgcn_wmma_f32_16x16x4_f32(false, t.a0[kk], false, t.b1[kk], (short)0, c01, false, false);
        c10 = __builtin_amdgcn_wmma_f32_16x16x4_f32(false, t.a1[kk], false, t.b0[kk], (short)0, c10, false, false);
        c11 = __builtin_amdgcn_wmma_f32_16x16x4_f32(false, t.a1[kk], false, t.b1[kk], (short)0, c11, false, false);
    }
}

// ---------------------------------------------------------------------------
// Fused GEMM:  Y[r, colOff+c] = act( X[r,:] . W[c,:] + bias[c] )
// One wave computes a 32x32 output tile (2x2 WMMA 16x16 sub-tiles), fp32
// WMMA 16x16x4, register double-buffered 16-k blocks (In % 32 == 0).
// PF: compile-time weight-prefetch for streaming-K layers.
// act: 0 = none, 1 = relu, 2 = leaky_relu(0.01)
// ---------------------------------------------------------------------------
template<bool PF>
__global__ __launch_bounds__(32) void gemm_wmma(
    const float* __restrict__ X, int ldX,
    const float* __restrict__ W, const float* __restrict__ Bias,
    float* __restrict__ Y, int ldY, int colOff, int In, int act)
{
    const int lane = threadIdx.x;
    const int half = lane >> 4;      // 0: lanes 0-15, 1: lanes 16-31
    const int l16  = lane & 15;
    const int row0 = blockIdx.x * 32;
    const int col0 = blockIdx.y * 32;

    // A operand: lane l16 = M row, K = k0 + 2*half + {0,1}  (16x4 f32 layout)
    const float* xr0 = X + (size_t)(row0 + l16) * ldX      + 2 * half;
    const float* xr1 = X + (size_t)(row0 + 16 + l16) * ldX + 2 * half;
    // B operand: lane l16 = N col, same K striping; B[k][n] = W[n][k]
    const float* wr0 = W + (size_t)(col0 + l16) * In       + 2 * half;
    const float* wr1 = W + (size_t)(col0 + 16 + l16) * In  + 2 * half;

    v8f c00 = {}, c01 = {}, c10 = {}, c11 = {};
    Blk tA, tB;

    load_blk(tA, xr0, xr1, wr0, wr1, 0);
    int k0 = 0;
    for (; k0 + 32 < In; k0 += 32) {
        if (PF) {   // branchless: clamp prefetch offset (uniform scalar select)
            int kp = (k0 + 64 < In) ? (k0 + 64) : k0;
            __builtin_prefetch(wr0 + kp, 0, 1);
            __builtin_prefetch(wr1 + kp, 0, 1);
        }
        load_blk(tB, xr0, xr1, wr0, wr1, k0 + 16);
        mma_blk(tA, c00, c01, c10, c11);
        load_blk(tA, xr0, xr1, wr0, wr1, k0 + 32);
        mma_blk(tB, c00, c01, c10, c11);
    }
    // tail: k0 + 32 == In
    load_blk(tB, xr0, xr1, wr0, wr1, k0 + 16);
    mma_blk(tA, c00, c01, c10, c11);
    mma_blk(tB, c00, c01, c10, c11);

    const float bv0 = Bias[col0 + l16];
    const float bv1 = Bias[col0 + 16 + l16];

    // C/D layout: element (M = v + 8*half, N = l16) per 16x16 sub-tile
#pragma unroll
    for (int v = 0; v < 8; v++) {
        int m = v + 8 * half;
        float v00 = c00[v] + bv0, v01 = c01[v] + bv1;
        float v10 = c10[v] + bv0, v11 = c11[v] + bv1;
        if (act == 1) {
            v00 = fmaxf(v00, 0.f); v01 = fmaxf(v01, 0.f);
            v10 = fmaxf(v10, 0.f); v11 = fmaxf(v11, 0.f);
        } else if (act == 2) {
            v00 = (v00 >= 0.f) ? v00 : 0.01f * v00;
            v01 = (v01 >= 0.f) ? v01 : 0.01f * v01;
            v10 = (v10 >= 0.f) ? v10 : 0.01f * v10;
            v11 = (v11 >= 0.f) ? v11 : 0.01f * v11;
        }
        float* yr0 = Y + (size_t)(row0 + m) * ldY + colOff + col0;
        float* yr1 = Y + (size_t)(row0 + 16 + m) * ldY + colOff + col0;
        yr0[l16]      = v00;
        yr0[16 + l16] = v01;
        yr1[l16]      = v10;
        yr1[16 + l16] = v11;
    }
}

// ---------------------------------------------------------------------------
// Small elementwise / reduction kernels
// ---------------------------------------------------------------------------
__global__ void residual_kernel(const float* __restrict__ a,
                                const float* __restrict__ b,
                                float* __restrict__ c, int n)
{
    int i = blockIdx.x * 256 + threadIdx.x;
    if (i < n) c[i] = a[i] + b[i];
}

// mean over K: X (B,K,D) -> M (B,D).  grid = B, block = D
__global__ void mean_k_kernel(const float* __restrict__ X,
                              float* __restrict__ M, int D)
{
    int b = blockIdx.x, d = threadIdx.x;
    float s = 0.f;
    for (int k = 0; k < K_; k++) s += X[((size_t)b * K_ + k) * D + d];
    M[(size_t)b * D + d] = s * (1.f / K_);
}

// mean over K + center: HX (B,K,32) -> MHX (B,32), HA = HX - MHX. grid B, block 32
__global__ void mean_center_kernel(const float* __restrict__ HX,
                                   float* __restrict__ MHX,
                                   float* __restrict__ HA)
{
    int b = blockIdx.x, d = threadIdx.x;
    float s = 0.f;
    for (int k = 0; k < K_; k++) s += HX[((size_t)b * K_ + k) * DZ_ + d];
    float m = s * (1.f / K_);
    MHX[b * DZ_ + d] = m;
    for (int k = 0; k < K_; k++) {
        size_t i = ((size_t)b * K_ + k) * DZ_ + d;
        HA[i] = HX[i] - m;
    }
}

// masked mc weight (padded to In=32) and padded xs rows
__global__ void prep_kernel(const float* __restrict__ mc_w,
                            const float* __restrict__ mask,
                            const float* __restrict__ raw,
                            float* __restrict__ MCW,
                            float* __restrict__ XSP)
{
    int i = blockIdx.x * 256 + threadIdx.x;
    if (i < 128 * 32) {
        int o = i >> 5, k = i & 31;
        MCW[i] = (k < RAW_) ? mc_w[o * RAW_ + k] * mask[k * 128 + o] : 0.f;
    }
    if (i < B_ * 32) {
        int b = i >> 5, k = i & 31;
        XSP[i] = (k < RAW_) ? raw[b * RAW_ + k] : 0.f;
    }
}

// d = (x + 0.001)^2 + sqrt(0.05^2 - 0.001)
__global__ void ddiag_kernel(const float* __restrict__ x,
                             float* __restrict__ d, int n)
{
    int i = blockIdx.x * 256 + threadIdx.x;
    if (i < n) {
        float t = x[i] + 0.001f;
        d[i] = t * t + 0.03872983346207417f;   // sqrt(0.0015)
    }
}

// scatter small outputs: obs broadcast, obs_z, H_X_mean, enc_out, m_state_pred
__global__ void scatter_kernel(const float* __restrict__ OBSB,
                               const float* __restrict__ ENC,
                               const float* __restrict__ MHX,
                               const float* __restrict__ MA,
                               float* __restrict__ out,
                               size_t off_obs, size_t off_obsz,
                               size_t off_hxm, size_t off_enc, size_t off_msp)
{
    int i = blockIdx.x * 256 + threadIdx.x;
    if (i < B_ * K_ * DZ_) {                  // obs: broadcast per-batch row over K
        int b = i / (K_ * DZ_);
        int d = i & (DZ_ - 1);
        out[off_obs + i] = OBSB[b * DZ_ + d];
    }
    if (i < B_ * DZ_) {
        out[off_obsz + i] = OBSB[i];
        out[off_hxm + i]  = MHX[i];
    }
    if (i < B_ * 64) {
        out[off_enc + i] = ENC[i];
        out[off_msp + i] = MA[i];
    }
}

// ---------------------------------------------------------------------------
// Per-batch Kalman update: innovation, 32x32 Gauss-Jordan inverse, gain,
// state_new, m_state_new.  grid = B, block = 256, all work in LDS.
// ---------------------------------------------------------------------------
__global__ __launch_bounds__(256) void kalman_kernel(
    const float* __restrict__ SP,   // state_pred (B,K,64)
    const float* __restrict__ HX,   // (B,K,32)
    const float* __restrict__ HA,   // (B,K,32)
    const float* __restrict__ MA,   // (B,64)
    const float* __restrict__ OBSB, // (B,32)
    const float* __restrict__ DD,   // (B,32) R diagonal
    float* __restrict__ out_state_new,
    float* __restrict__ out_mstate_new)
{
    __shared__ float sHA[K_ * DZ_];      // [k][d]
    __shared__ float sM[DZ_ * K_];       // [d][k] = obs - H_X^T
    __shared__ float sAug[DZ_][66];      // Gauss-Jordan augmented (pad 66)
    __shared__ float sSP[K_ * DX_];      // [k][x]
    __shared__ float sAH[DX_ * DZ_];     // A@H_A, later gain^T
    __shared__ float sKG[DX_ * DZ_];     // Kalman gain
    __shared__ float sMA[DX_];
    __shared__ float sFac[DZ_];

    const int b = blockIdx.x, tid = threadIdx.x;
    const float invfac = 1.f / (K_ - 1);

    for (int i = tid; i < K_ * DZ_; i += 256) sHA[i] = HA[(size_t)b * K_ * DZ_ + i];
    for (int i = tid; i < K_ * DX_; i += 256) sSP[i] = SP[(size_t)b * K_ * DX_ + i];
    if (tid < DX_) sMA[tid] = MA[b * DX_ + tid];
    for (int i = tid; i < DZ_ * K_; i += 256) {
        int d = i >> 5, k = i & 31;
        sM[i] = OBSB[b * DZ_ + d] - HX[((size_t)b * K_ + k) * DZ_ + d];
    }
    __syncthreads();

    // innovation = invfac * H_A^T H_A + diag(d); augmented identity
    for (int i = tid; i < DZ_ * DZ_; i += 256) {
        int r = i >> 5, c = i & 31;
        float s = 0.f;
        for (int k = 0; k < K_; k++) s += sHA[k * DZ_ + r] * sHA[k * DZ_ + c];
        sAug[r][c] = invfac * s + ((r == c) ? DD[b * DZ_ + r] : 0.f);
        sAug[r][32 + c] = (r == c) ? 1.f : 0.f;
    }
    __syncthreads();

    // Gauss-Jordan (SPD + positive diagonal -> no pivoting needed)
    for (int p = 0; p < DZ_; p++) {
        if (tid < DZ_) sFac[tid] = sAug[tid][p];
        __syncthreads();
        float pinv = 1.f / sFac[p];
        if (tid < 64) sAug[p][tid] *= pinv;
        __syncthreads();
        for (int i = tid; i < DZ_ * 64; i += 256) {
            int r = i >> 6, c = i & 63;
            if (r != p) sAug[r][c] -= sFac[r] * sAug[p][c];
        }
        __syncthreads();
    }

    // AH[x][d] = sum_k (SP[k][x]-MA[x]) * HA[k][d]
    for (int i = tid; i < DX_ * DZ_; i += 256) {
        int x = i >> 5, d = i & 31;
        float s = 0.f, mx = sMA[x];
        for (int k = 0; k < K_; k++) s += (sSP[k * DX_ + x] - mx) * sHA[k * DZ_ + d];
        sAH[i] = s;
    }
    __syncthreads();

    // Kg[x][d] = invfac * sum_j AH[x][j] * inv[j][d]
    for (int i = tid; i < DX_ * DZ_; i += 256) {
        int x = i >> 5, d = i & 31;
        float s = 0.f;
        for (int j = 0; j < DZ_; j++) s += sAH[x * DZ_ + j] * sAug[j][32 + d];
        sKG[i] = invfac * s;
    }
    __syncthreads();

    // gain^T[x][k] = sum_d Kg[x][d] * M[d][k]   (reuse sAH)
    for (int i = tid; i < DX_ * K_; i += 256) {
        int x = i >> 5, k = i & 31;
        float s = 0.f;
        for (int d = 0; d < DZ_; d++) s += sKG[x * DZ_ + d] * sM[d * K_ + k];
        sAH[i] = s;
    }
    __syncthreads();

    // state_new[k][x] = SP[k][x] + gain^T[x][k]
    for (int i = tid; i < K_ * DX_; i += 256) {
        int k = i >> 6, x = i & 63;
        out_state_new[(size_t)b * K_ * DX_ + i] = sSP[i] + sAH[x * K_ + k];
    }
    if (tid < DX_) {
        float s = 0.f;
        for (int k = 0; k < K_; k++) s += sSP[k * DX_ + tid] + sAH[tid * K_ + k];
        out_mstate_new[b * DX_ + tid] = s * (1.f / K_);
    }
}

// ---------------------------------------------------------------------------
// Host launcher
// ---------------------------------------------------------------------------
extern "C" void kernel_launch(void* const* d_in, const int* in_sizes, int n_in,
                              void* d_out, int out_size, void* d_ws, size_t ws_size,
                              hipStream_t stream)
{
    const float* action    = (const float*)d_in[0];
    const float* raw_obs   = (const float*)d_in[1];
    const float* state_old = (const float*)d_in[2];
    /* m_state d_in[3] unused by reference */
    const float* mask      = (const float*)d_in[4];
    // params: _LAYERS order, (w, b) pairs starting at d_in[5]
    auto W = [&](int layer) { return (const float*)d_in[5 + 2 * layer]; };
    auto Bv = [&](int layer) { return (const float*)d_in[5 + 2 * layer + 1]; };
    enum { L_mc, L_s2, L_s3, L_s4, L_s5, L_s6, L_o1, L_o2, L_o3, L_o4, L_o5,
           L_n1, L_n2, L_p1, L_p2, L_p3, L_pa1, L_pa2, L_pa3, L_p4, L_p5 };

    float* ws = (float*)d_ws;
    // workspace layout (floats)
    float* SP   = ws;                          // 16384*64
    float* HX   = SP   + (size_t)BK_ * 64;     // 16384*32
    float* HA   = HX   + (size_t)BK_ * 32;     // 16384*32
    float* MA   = HA   + (size_t)BK_ * 32;     // 512*64
    float* MHX  = MA   + (size_t)B_ * 64;      // 512*32
    float* T0   = MHX  + (size_t)B_ * 32;      // 16384*128
    float* T1   = T0   + (size_t)BK_ * 128;    // 16384*128
    float* T2   = T1   + (size_t)BK_ * 128;    // 16384*128
    float* ENC  = T2   + (size_t)BK_ * 128;    // 512*64
    float* OBSB = ENC  + (size_t)B_ * 64;      // 512*32
    float* DD   = OBSB + (size_t)B_ * 32;      // 512*32
    float* MCW  = DD   + (size_t)B_ * 32;      // 128*32
    float* XSP  = MCW  + (size_t)128 * 32;     // 512*32

    float* out = (float*)d_out;
    const size_t OUT_SN   = 0;                           // state_new  (B,K,64)
    const size_t OUT_MSN  = OUT_SN  + (size_t)BK_ * 64;  // m_state_new (B,1,64)
    const size_t OUT_MSP  = OUT_MSN + (size_t)B_ * 64;   // m_state_pred
    const size_t OUT_OBSZ = OUT_MSP + (size_t)B_ * 64;   // obs_z (B,1,32)
    const size_t OUT_OBS  = OUT_OBSZ + (size_t)B_ * 32;  // obs (B,K,32)
    const size_t OUT_HXM  = OUT_OBS + (size_t)BK_ * 32;  // H_X_mean
    const size_t OUT_ENC  = OUT_HXM + (size_t)B_ * 32;   // enc_out (B,64)

    auto gemm = [&](const float* X, int ldX, int layer, float* Y, int ldY,
                    int colOff, int R, int In, int Out, int act) {
        dim3 g(R / 32, Out / 32);
        if (In >= 512)
            gemm_wmma<true><<<g, 32, 0, stream>>>(X, ldX, W(layer), Bv(layer),
                                                  Y, ldY, colOff, In, act);
        else
            gemm_wmma<false><<<g, 32, 0, stream>>>(X, ldX, W(layer), Bv(layer),
                                                   Y, ldY, colOff, In, act);
    };

    // --- prep: masked mc weight (pad In 30->32), padded xs rows ---
    prep_kernel<<<(B_ * 32 + 255) / 256, 256, 0, stream>>>(W(L_mc), mask, raw_obs, MCW, XSP);

    // --- p-branch (B*K rows) ---
    gemm(state_old, DX_, L_p1, T0, 64, 0, BK_, 64, 64, 1);
    gemm(T0, 64,  L_p2, T1, 128, 0,  BK_, 64, 128, 1);
    gemm(T1, 128, L_p3, T2, 128, 0,  BK_, 128, 64, 1);   // merge[:, 0:64]
    gemm(action, DA_, L_pa1, T0, 64, 0, BK_, 32, 64, 1);
    gemm(T0, 64,  L_pa2, T1, 128, 0, BK_, 64, 128, 1);
    gemm(T1, 128, L_pa3, T2, 128, 64, BK_, 128, 64, 1);  // merge[:, 64:128]
    gemm(T2, 128, L_p4, T0, 64, 0, BK_, 128, 64, 0);
    gemm(T0, 64,  L_p5, T1, 64, 0, BK_, 64, 64, 0);      // update
    residual_kernel<<<(BK_ * 64 + 255) / 256, 256, 0, stream>>>(state_old, T1, SP, BK_ * 64);
    mean_k_kernel<<<B_, 64, 0, stream>>>(SP, MA, 64);    // m_A (= m_state_pred)

    // --- o-branch (B*K rows) ---
    gemm(SP, 64,  L_o1, T0, 64, 0, BK_, 64, 64, 1);
    gemm(T0, 64,  L_o2, T1, 128, 0, BK_, 64, 128, 1);
    gemm(T1, 128, L_o3, T0, 128, 0, BK_, 128, 128, 1);
    gemm(T0, 128, L_o4, T1, 64, 0, BK_, 128, 64, 1);
    gemm(T1, 64,  L_o5, HX, 32, 0, BK_, 64, 32, 0);      // H_X
    mean_center_kernel<<<B_, 32, 0, stream>>>(HX, MHX, HA);

    // --- s-branch: raw_obs is broadcast over K, so compute on B rows only ---
    {   // first layer uses the premultiplied/padded MCW weight
        dim3 g(B_ / 32, 128 / 32);
        gemm_wmma<false><<<g, 32, 0, stream>>>(XSP, 32, MCW, Bv(L_mc), T0, 128, 0, 32, 2);
    }
    gemm(T0, 128,  L_s2, T1, 512, 0,  B_, 128, 512, 2);
    gemm(T1, 512,  L_s3, T0, 1024, 0, B_, 512, 1024, 2);
    gemm(T0, 1024, L_s4, T1, 2048, 0, B_, 1024, 2048, 2);
    gemm(T1, 2048, L_s5, ENC, 64, 0,  B_, 2048, 64, 2);  // encoding rows
    gemm(ENC, 64,  L_s6, OBSB, 32, 0, B_, 64, 32, 0);    // obs rows

    // --- noise net (on encoding, per-batch) ---
    gemm(ENC, 64, L_n1, T0, 32, 0, B_, 64, 32, 1);
    gemm(T0, 32,  L_n2, T1, 32, 0, B_, 32, 32, 0);
    ddiag_kernel<<<(B_ * 32 + 255) / 256, 256, 0, stream>>>(T1, DD, B_ * 32);

    // --- small outputs ---
    scatter_kernel<<<(BK_ * DZ_ + 255) / 256, 256, 0, stream>>>(
        OBSB, ENC, MHX, MA, out, OUT_OBS, OUT_OBSZ, OUT_HXM, OUT_ENC, OUT_MSP);

    // --- Kalman update per batch ---
    kalman_kernel<<<B_, 256, 0, stream>>>(SP, HX, HA, MA, OBSB, DD,
                                          out + OUT_SN, out + OUT_MSN);
}